// BiLSTM_Parser_65189013619009
// MI455X (gfx1250) — compile-verified
//
#include <hip/hip_runtime.h>
#include <math.h>

typedef __attribute__((ext_vector_type(2))) float v2f;
typedef __attribute__((ext_vector_type(8))) float v8f;

#define SEQ 256
#define HID 512
#define G4  2048
#define EMBD 400
#define LHID 1024
#define MLPD 1024

#if __has_builtin(__builtin_amdgcn_tanhf)
__device__ __forceinline__ float fast_tanh(float x) { return __builtin_amdgcn_tanhf(x); }
#else
__device__ __forceinline__ float fast_tanh(float x) { return tanhf(x); }
#endif
__device__ __forceinline__ float fast_sig(float x) { return 0.5f * fast_tanh(0.5f * x) + 0.5f; }

// ---------------------------------------------------------------------------
// Embedding gather: emb[t] = concat(word_table[word[t]], pos_table[pos[t]])
// ---------------------------------------------------------------------------
__global__ void embed_kernel(const int* __restrict__ word,
                             const int* __restrict__ pos,
                             const float* __restrict__ wtab,
                             const float* __restrict__ ptab,
                             float* __restrict__ emb) {
  int t = blockIdx.x;
  const float* wr = wtab + (size_t)word[t] * 300;
  const float* pr = ptab + (size_t)pos[t] * 100;
  float* er = emb + (size_t)t * EMBD;
  for (int k = threadIdx.x; k < 300; k += blockDim.x) er[k] = wr[k];
  for (int k = threadIdx.x; k < 100; k += blockDim.x) er[300 + k] = pr[k];
}

// ---------------------------------------------------------------------------
// f32 WMMA GEMM: out[m][n] = sum_k X[m][k] * W[n][k] (+ bias0[n] + bias1[n])
// One wave computes one 16x16 tile via V_WMMA_F32_16X16X4_F32 (exact fp32).
// M, N multiples of 16; K multiple of 4. Tile count must divide grid*8.
// ---------------------------------------------------------------------------
__global__ void wmma_gemm_f32(const float* __restrict__ X, int ldx,
                              const float* __restrict__ W, int ldw,
                              const float* __restrict__ bias0,
                              const float* __restrict__ bias1,
                              float* __restrict__ out, int ldo,
                              int M, int N, int K) {
  int wave  = blockIdx.x * (blockDim.x >> 5) + (threadIdx.x >> 5);
  int tilesN = N >> 4;
  int tm = wave / tilesN;
  int tn = wave % tilesN;
  if (tm * 16 >= M) return;                 // wave-uniform: EXEC stays all-1s
  int lane  = threadIdx.x & 31;
  int l15   = lane & 15;
  int khalf = (lane >> 4) * 2;              // lanes 0-15: K={0,1}; 16-31: K={2,3}

  const float* xrow = X + (size_t)(tm * 16 + l15) * ldx;   // A row (M index)
  const float* wrow = W + (size_t)(tn * 16 + l15) * ldw;   // B col (N index)

  v8f c = {};
  #pragma unroll 4
  for (int k = 0; k < K; k += 4) {
    v2f a = *(const v2f*)(xrow + k + khalf);
    v2f b = *(const v2f*)(wrow + k + khalf);
    c = __builtin_amdgcn_wmma_f32_16x16x4_f32(false, a, false, b,
                                              (short)0, c, false, false);
  }

  int col = tn * 16 + l15;
  float bias = 0.0f;
  if (bias0) bias += bias0[col];
  if (bias1) bias += bias1[col];
  int rbase = tm * 16 + ((lane >> 4) << 3); // lanes 16-31 hold rows M=8..15
  #pragma unroll
  for (int r = 0; r < 8; ++r)
    out[(size_t)(rbase + r) * ldo + col] = c[r] + bias;
}

// ---------------------------------------------------------------------------
// LSTM recurrence, one direction per block (blockIdx.x = dir; dir 1 = reverse)
// g = gx[t] + h @ w_hh.T ; gates -> (h, c); h kept in LDS; 1024 threads,
// each thread owns gate rows (tid, tid+1024).
// ---------------------------------------------------------------------------
__global__ void __launch_bounds__(1024)
lstm_kernel(const float* __restrict__ gx_base,
            const float* __restrict__ whh_base,
            float* __restrict__ hout) {
  const int dir = blockIdx.x;
  const float* gx   = gx_base  + (size_t)dir * SEQ * G4;
  const float* w_hh = whh_base + (size_t)dir * G4 * HID;
  const int col_off = dir * HID;
  const int rev = dir;

  __shared__ __align__(16) float h_s[HID];
  __shared__ float g_s[G4];

  const int tid = threadIdx.x;
  float c_reg = 0.0f;
  if (tid < HID) h_s[tid] = 0.0f;
  __syncthreads();

  const float4* w0 = (const float4*)(w_hh + (size_t)tid * HID);
  const float4* w1 = (const float4*)(w_hh + (size_t)(tid + 1024) * HID);
  const float4* h4 = (const float4*)h_s;

  for (int s = 0; s < SEQ; ++s) {
    const int t = rev ? (SEQ - 1 - s) : s;
    const float* grow = gx + (size_t)t * G4;
    if (s + 1 < SEQ) {                      // global_prefetch_b8: next pre-gates row
      const int tn = rev ? (SEQ - 2 - s) : (s + 1);
      __builtin_prefetch(gx + (size_t)tn * G4 + tid, 0, 0);
    }

    float acc0 = 0.0f, acc1 = 0.0f;
    #pragma unroll 4
    for (int k = 0; k < HID / 4; ++k) {
      float4 hv = h4[k];
      float4 a = w0[k];
      float4 b = w1[k];
      acc0 += a.x * hv.x + a.y * hv.y + a.z * hv.z + a.w * hv.w;
      acc1 += b.x * hv.x + b.y * hv.y + b.z * hv.z + b.w * hv.w;
    }
    g_s[tid]        = acc0 + grow[tid];
    g_s[tid + 1024] = acc1 + grow[tid + 1024];
    __syncthreads();

    if (tid < HID) {
      float gi = g_s[tid];
      float gf = g_s[tid + 512];
      float gg = g_s[tid + 1024];
      float go = g_s[tid + 1536];
      c_reg = fast_sig(gf) * c_reg + fast_sig(gi) * fast_tanh(gg);
      float h = fast_sig(go) * fast_tanh(c_reg);
      h_s[tid] = h;
      hout[(size_t)t * LHID + col_off + tid] = h;
    }
    __syncthreads();
  }
}

// ---------------------------------------------------------------------------
// scores[i][j] = b2 + sum_k tanh(A[i][k] + B[j][k] + b1[k]) * w2[k]; NaN diag.
// grid (256, 32), block 256 = 8 waves; one wave per j, A[i]+b1 staged in LDS.
// ---------------------------------------------------------------------------
__global__ void score_kernel(const float* __restrict__ A,
                             const float* __restrict__ B,
                             const float* __restrict__ b1,
                             const float* __restrict__ w2,
                             const float* __restrict__ b2,
                             float* __restrict__ out) {
  __shared__ float Ai[MLPD];
  const int i = blockIdx.x;
  for (int k = threadIdx.x; k < MLPD; k += blockDim.x)
    Ai[k] = A[(size_t)i * MLPD + k] + b1[k];
  __syncthreads();

  const int wave = threadIdx.x >> 5;
  const int lane = threadIdx.x & 31;
  const int j = blockIdx.y * 8 + wave;
  const float* Bj = B + (size_t)j * MLPD;

  float acc = 0.0f;
  #pragma unroll 8
  for (int k = lane; k < MLPD; k += 32)
    acc += fast_tanh(Ai[k] + Bj[k]) * w2[k];

  #pragma unroll
  for (int off = 16; off > 0; off >>= 1)
    acc += __shfl_xor(acc, off, 32);

  if (lane == 0)
    out[(size_t)i * SEQ + j] =
        (i == j) ? __uint_as_float(0x7fc00000u) : (acc + b2[0]);
}

// ---------------------------------------------------------------------------
extern "C" void kernel_launch(void* const* d_in, const int* in_sizes, int n_in,
                              void* d_out, int out_size, void* d_ws, size_t ws_size,
                              hipStream_t stream) {
  (void)in_sizes; (void)n_in; (void)out_size; (void)ws_size;
  const int*   word  = (const int*)d_in[0];
  const int*   pos   = (const int*)d_in[1];
  const float* wtab  = (const float*)d_in[2];
  const float* ptab  = (const float*)d_in[3];
  const float* w_ih0 = (const float*)d_in[4];
  const float* w_hh0 = (const float*)d_in[5];
  const float* b_ih0 = (const float*)d_in[6];
  const float* b_hh0 = (const float*)d_in[7];
  const float* w_ih1 = (const float*)d_in[8];
  const float* w_hh1 = (const float*)d_in[9];
  const float* b_ih1 = (const float*)d_in[10];
  const float* b_hh1 = (const float*)d_in[11];
  const float* W1    = (const float*)d_in[12];
  const float* b1    = (const float*)d_in[13];
  const float* w2    = (const float*)d_in[14];
  const float* b2    = (const float*)d_in[15];
  float* out = (float*)d_out;

  float* ws  = (float*)d_ws;
  float* emb = ws;                         // 256*400
  float* gx  = emb + SEQ * EMBD;           // 2*256*2048 (reused both layers)
  float* h0  = gx + 2 * SEQ * G4;          // 256*1024
  float* h1  = h0 + SEQ * LHID;            // 256*1024
  float* Ab  = h1 + SEQ * LHID;            // 256*1024
  float* Bb  = Ab + SEQ * MLPD;            // 256*1024

  embed_kernel<<<SEQ, 128, 0, stream>>>(word, pos, wtab, ptab, emb);

  // Layer 0 pre-gates: gx[d] = emb @ w_ih0[d].T + b_ih0[d] + b_hh0[d]
  for (int d = 0; d < 2; ++d)
    wmma_gemm_f32<<<256, 256, 0, stream>>>(
        emb, EMBD, w_ih0 + (size_t)d * G4 * EMBD, EMBD,
        b_ih0 + d * G4, b_hh0 + d * G4,
        gx + (size_t)d * SEQ * G4, G4, SEQ, G4, EMBD);

  lstm_kernel<<<2, 1024, 0, stream>>>(gx, w_hh0, h0);

  // Layer 1 pre-gates: gx[d] = h0 @ w_ih1[d].T + b_ih1[d] + b_hh1[d]
  for (int d = 0; d < 2; ++d)
    wmma_gemm_f32<<<256, 256, 0, stream>>>(
        h0, LHID, w_ih1 + (size_t)d * G4 * LHID, LHID,
        b_ih1 + d * G4, b_hh1 + d * G4,
        gx + (size_t)d * SEQ * G4, G4, SEQ, G4, LHID);

  lstm_kernel<<<2, 1024, 0, stream>>>(gx, w_hh1, h1);

  // Biaffine projections: A = h1 @ Wl.T, B = h1 @ Wr.T (Wl/Wr are W1 halves)
  wmma_gemm_f32<<<128, 256, 0, stream>>>(h1, LHID, W1, 2 * LHID,
                                         nullptr, nullptr, Ab, MLPD,
                                         SEQ, MLPD, LHID);
  wmma_gemm_f32<<<128, 256, 0, stream>>>(h1, LHID, W1 + LHID, 2 * LHID,
                                         nullptr, nullptr, Bb, MLPD,
                                         SEQ, MLPD, LHID);

  score_kernel<<<dim3(SEQ, SEQ / 8), 256, 0, stream>>>(Ab, Bb, b1, w2, b2, out);
}